// GINModel_69131793596458
// MI455X (gfx1250) — compile-verified
//
#include <hip/hip_runtime.h>
#include <hip/hip_bf16.h>

// ---------------------------------------------------------------------------
// GIN (3x GINConv + MLP head), N=100000 nodes, E=1.6M edges, H=48, DOUT=16.
// Memory-bound on the scatter-sum aggregation (~1 GB of gather+atomic traffic
// across 3 layers -> ~50-100us at 23.3 TB/s). GEMMs are tiny (3.4 GFLOP) so
// we keep them in exact fp32 using V_WMMA_F32_16X16X4_F32.
// ---------------------------------------------------------------------------

typedef __attribute__((ext_vector_type(2))) float v2f;
typedef __attribute__((ext_vector_type(8))) float v8f;

#define NNODES 100000
#define NFEAT  48

// ---------------------------------------------------------------------------
// Zero a float buffer (workspace is poisoned by the harness).
// ---------------------------------------------------------------------------
__global__ void zero_f32(float* __restrict__ p, int n) {
    int i = blockIdx.x * blockDim.x + threadIdx.x;
    if (i < n) p[i] = 0.0f;
}

// ---------------------------------------------------------------------------
// Scatter-sum: agg[dst[e], f] += h[src[e], f]  for all edges, f in [0,48).
// 16 threads per edge, 3 features per thread. atomicAdd -> global_atomic_add_f32.
// ---------------------------------------------------------------------------
__global__ void scatter_add48(const float* __restrict__ h,
                              const int* __restrict__ src,
                              const int* __restrict__ dst,
                              float* __restrict__ agg, int E) {
    int t = blockIdx.x * blockDim.x + threadIdx.x;
    int e = t >> 4;
    if (e >= E) return;
    int f0 = (t & 15) * 3;
    const float* hs = h + (size_t)src[e] * NFEAT + f0;
    float*       ad = agg + (size_t)dst[e] * NFEAT + f0;
    atomicAdd(&ad[0], hs[0]);
    atomicAdd(&ad[1], hs[1]);
    atomicAdd(&ad[2], hs[2]);
}

// ---------------------------------------------------------------------------
// Fused GEMM: out[N x DOUT] = act( (in [+ in2]) @ W[48 x DOUT] + b )
// One wave computes a 16-row x DOUT tile with V_WMMA_F32_16X16X4_F32,
// K-loop of 12 steps (48/4). 256-thread block = 8 waves = 128 rows.
//
// fp32 16x16x4 fragment layouts (ISA 7.12.2):
//   A (16x4): lanes 0-15 hold M=lane, {K0,K1}; lanes 16-31 hold M=lane-16, {K2,K3}
//   B (4x16): lanes 0-15 hold N=lane, {K0,K1}; lanes 16-31 hold N=lane-16, {K2,K3}
//   C/D (16x16): 8 VGPRs; lane<16 -> M=v, N=lane; lane>=16 -> M=v+8, N=lane-16
// ---------------------------------------------------------------------------
template <int DOUT, bool RELU, bool HASADD>
__global__ void gemm48(const float* __restrict__ in,
                       const float* __restrict__ in2,
                       const float* __restrict__ W,
                       const float* __restrict__ bias,
                       float* __restrict__ out, int n) {
    constexpr int DIN = 48;
    constexpr int NT  = DOUT / 16;   // 16-col output tiles

    __shared__ float sW[DIN * DOUT];
    __shared__ float sB[DOUT];

    int tid = threadIdx.x;
    for (int i = tid; i < DIN * DOUT; i += blockDim.x) sW[i] = W[i];
    if (tid < DOUT) sB[tid] = bias[tid];
    __syncthreads();

    int wave    = tid >> 5;
    int lane    = tid & 31;
    int rowTile = blockIdx.x * 8 + wave;
    int row0    = rowTile * 16;
    if (row0 >= n) return;          // whole-wave exit: EXEC stays all-ones

    int m     = lane & 15;          // row (A) / col (B,D) within tile
    int khalf = (lane >> 4) * 2;    // 0 for lanes 0-15, 2 for lanes 16-31
    int node  = row0 + m;           // n is a multiple of 16 (100000 = 6250*16)

    const float* arow  = in + (size_t)node * DIN;
    const float* arow2 = HASADD ? (in2 + (size_t)node * DIN) : nullptr;

    v8f acc[NT] = {};

#pragma unroll
    for (int kk = 0; kk < DIN; kk += 4) {
        int k0 = kk + khalf;
        v2f a;
        a.x = arow[k0];
        a.y = arow[k0 + 1];
        if (HASADD) {               // fused GIN residual: (1+eps)*x + agg, eps=0
            a.x += arow2[k0];
            a.y += arow2[k0 + 1];
        }
#pragma unroll
        for (int t = 0; t < NT; t++) {
            int col = t * 16 + m;
            v2f b;
            b.x = sW[k0 * DOUT + col];
            b.y = sW[(k0 + 1) * DOUT + col];
            acc[t] = __builtin_amdgcn_wmma_f32_16x16x4_f32(
                false, a, false, b, (short)0, acc[t], false, false);
        }
    }

    // Bias + activation + writeback (lane<16 -> rows v, lane>=16 -> rows v+8).
    int mbase = (lane < 16) ? 0 : 8;
#pragma unroll
    for (int t = 0; t < NT; t++) {
        int col = t * 16 + m;
        float bv = sB[col];
#pragma unroll
        for (int v = 0; v < 8; v++) {
            float val = acc[t][v] + bv;
            if (RELU) val = fmaxf(val, 0.0f);
            out[(size_t)(row0 + mbase + v) * DOUT + col] = val;
        }
    }
}

// ---------------------------------------------------------------------------
// Orchestration
// ---------------------------------------------------------------------------
extern "C" void kernel_launch(void* const* d_in, const int* in_sizes, int n_in,
                              void* d_out, int out_size, void* d_ws, size_t ws_size,
                              hipStream_t stream) {
    const float* x   = (const float*)d_in[0];
    const int*   ei  = (const int*)d_in[1];
    const int    E   = in_sizes[1] / 2;
    const int    n   = in_sizes[0] / NFEAT;   // 100000
    const int*   src = ei;
    const int*   dst = ei + E;

    const float* w11 = (const float*)d_in[2],  *b11 = (const float*)d_in[3];
    const float* w12 = (const float*)d_in[4],  *b12 = (const float*)d_in[5];
    const float* w21 = (const float*)d_in[6],  *b21 = (const float*)d_in[7];
    const float* w22 = (const float*)d_in[8],  *b22 = (const float*)d_in[9];
    const float* w31 = (const float*)d_in[10], *b31 = (const float*)d_in[11];
    const float* w32 = (const float*)d_in[12], *b32 = (const float*)d_in[13];
    const float* wf1 = (const float*)d_in[14], *bf1 = (const float*)d_in[15];
    const float* wf2 = (const float*)d_in[16], *bf2 = (const float*)d_in[17];

    float* out = (float*)d_out;

    // Workspace: 3 x N*48 float buffers (19.2 MB each; L2-resident on 192 MB L2)
    float* bufA = (float*)d_ws;
    float* bufB = bufA + (size_t)n * NFEAT;
    float* agg  = bufB + (size_t)n * NFEAT;

    const int nh       = n * NFEAT;
    const int zgrid    = (nh + 255) / 256;
    const int sgrid    = (E * 16 + 255) / 256;
    const int rowTiles = (n + 15) / 16;                // 6250
    const int ggrid    = (rowTiles + 7) / 8;           // 8 waves per block

    // ---- Layer 1: agg = scatter(x); h = relu((x+agg)@w11+b11); h = relu(h@w12+b12)
    zero_f32<<<zgrid, 256, 0, stream>>>(agg, nh);
    scatter_add48<<<sgrid, 256, 0, stream>>>(x, src, dst, agg, E);
    gemm48<48, true, true ><<<ggrid, 256, 0, stream>>>(x,    agg, w11, b11, bufA, n);
    gemm48<48, true, false><<<ggrid, 256, 0, stream>>>(bufA, nullptr, w12, b12, bufB, n);

    // ---- Layer 2
    zero_f32<<<zgrid, 256, 0, stream>>>(agg, nh);
    scatter_add48<<<sgrid, 256, 0, stream>>>(bufB, src, dst, agg, E);
    gemm48<48, true, true ><<<ggrid, 256, 0, stream>>>(bufB, agg, w21, b21, bufA, n);
    gemm48<48, true, false><<<ggrid, 256, 0, stream>>>(bufA, nullptr, w22, b22, bufB, n);

    // ---- Layer 3
    zero_f32<<<zgrid, 256, 0, stream>>>(agg, nh);
    scatter_add48<<<sgrid, 256, 0, stream>>>(bufB, src, dst, agg, E);
    gemm48<48, true, true ><<<ggrid, 256, 0, stream>>>(bufB, agg, w31, b31, bufA, n);
    gemm48<48, true, false><<<ggrid, 256, 0, stream>>>(bufA, nullptr, w32, b32, bufB, n);

    // ---- Head: relu(h@wf1+bf1) @ wf2 + bf2  -> [N x 16]
    gemm48<48, true, false><<<ggrid, 256, 0, stream>>>(bufB, nullptr, wf1, bf1, bufA, n);
    gemm48<16, false, false><<<ggrid, 256, 0, stream>>>(bufA, nullptr, wf2, bf2, out, n);
}